// CustomTransformerModel_71734543778611
// MI455X (gfx1250) — compile-verified
//
#include <hip/hip_runtime.h>
#include <hip/hip_bf16.h>

// ---------------------------------------------------------------------------
// CDNA5 (gfx1250) implementation.  All GEMMs run on V_WMMA_F32_16X16X4_F32
// (fp32 in / fp32 accumulate, wave32) with LDS-staged 64x64x32 block tiles
// and float4 (b128) staging when K % 4 == 0.  Small head-dim-5 attention and
// the LayerNorms run on VALU with wave32 shuffle reductions.
// ---------------------------------------------------------------------------

typedef float v2f __attribute__((ext_vector_type(2)));
typedef float v4f __attribute__((ext_vector_type(4)));
typedef float v8f __attribute__((ext_vector_type(8)));

#define LN_EPS 1e-5f
#define LDS_STRIDE 36  // even (8B-aligned b64 frags), mult of 4 (16B b128 staging),
                       // r*36 mod 64 distinct for r=0..15 -> conflict-free frags

// ---------------------------------------------------------------------------
// WMMA GEMM:  C[M,N] = act( A[M,K] @ W[N,K]^T + bias[N] )
// Block = 256 threads = 8 waves arranged 4 (M) x 2 (N); each wave owns a
// 16(M) x 32(N) tile (two accumulators sharing one A fragment).
// Block tile 64x64, K-blocked by 32 through LDS.  All loads are branchless:
//   - M/N edges: clamp row index (OOB A rows only feed OOB D rows, OOB W rows
//     only feed OOB D cols; both are masked at the store).
//   - K tail: zero-select on the W tile only (0 * garbage-A == 0).
// Fragment layouts per ISA 7.12.2:
//   A 16x4: lanes 0-15 hold {K=k,k+1}, lanes 16-31 hold {K=k+2,k+3}
//   D 16x16: VGPR j -> M=j (lanes 0-15) / M=j+8 (lanes 16-31), N=lane%16
// ---------------------------------------------------------------------------
template <int RELU, int VEC>
__global__ __launch_bounds__(256) void wmma_gemm_kernel(
    const float* __restrict__ A, const float* __restrict__ W,
    const float* __restrict__ bias, float* __restrict__ C,
    int M, int N, int K) {
  __shared__ float As[64 * LDS_STRIDE];
  __shared__ float Ws[64 * LDS_STRIDE];
  const int tid  = threadIdx.x;
  const int wave = tid >> 5;
  const int lane = tid & 31;
  const int wm = wave >> 1;          // 0..3
  const int wn = wave & 1;           // 0..1
  const int blockM = blockIdx.x * 64;
  const int blockN = blockIdx.y * 64;
  const int half = lane >> 4;        // 0: K pair {0,1}, 1: {2,3}
  const int idx  = lane & 15;

  v8f acc0 = {};
  v8f acc1 = {};
  for (int k0 = 0; k0 < K; k0 += 32) {
    __syncthreads();  // previous iteration's readers done before overwrite
    if (VEC) {
      // float4 staging: tile 64x32 = 512 groups of 4; 2 groups per thread.
#pragma unroll
      for (int i = 0; i < 2; ++i) {
        const int g = tid + i * 256;
        const int r = g >> 3, c4 = (g & 7) * 4;   // consecutive lanes -> coalesced
        {
          const int gm = min(blockM + r, M - 1);
          const int gk = min(k0 + c4, K - 4);     // K % 4 == 0 in VEC mode
          v4f av = *(const v4f*)(A + (size_t)gm * (size_t)K + gk);
          *(v4f*)(As + r * LDS_STRIDE + c4) = av; // garbage-K ok: W side zeroed
        }
        {
          const int gn = min(blockN + r, N - 1);
          const int gk = min(k0 + c4, K - 4);
          v4f wv = *(const v4f*)(W + (size_t)gn * (size_t)K + gk);
          if (k0 + c4 >= K) { wv.x = 0.f; wv.y = 0.f; wv.z = 0.f; wv.w = 0.f; }
          *(v4f*)(Ws + r * LDS_STRIDE + c4) = wv;
        }
      }
    } else {
      // scalar staging (K not a multiple of 4; only the K=25 channel GEMMs)
#pragma unroll
      for (int i = 0; i < 8; ++i) {
        const int e = tid + i * 256;
        const int r = e >> 5, c = e & 31;
        const int gm = min(blockM + r, M - 1);
        const int gk = min(k0 + c, K - 1);
        As[r * LDS_STRIDE + c] = A[(size_t)gm * (size_t)K + gk];
      }
#pragma unroll
      for (int i = 0; i < 8; ++i) {
        const int e = tid + i * 256;
        const int r = e >> 5, c = e & 31;
        const int gn = min(blockN + r, N - 1);
        const int gk = min(k0 + c, K - 1);
        float v = W[(size_t)gn * (size_t)K + gk];
        if (k0 + c >= K) v = 0.0f;               // branchless K-tail zero pad
        Ws[r * LDS_STRIDE + c] = v;
      }
    }
    __syncthreads();
    const float* ap  = As + (wm * 16 + idx) * LDS_STRIDE + 2 * half;
    const float* wp0 = Ws + (wn * 32 + idx) * LDS_STRIDE + 2 * half;
    const float* wp1 = wp0 + 16 * LDS_STRIDE;
#pragma unroll
    for (int kk = 0; kk < 32; kk += 4) {
      v2f a, b0, b1;
      a.x  = ap[kk];  a.y  = ap[kk + 1];         // ds_load_b64
      b0.x = wp0[kk]; b0.y = wp0[kk + 1];
      b1.x = wp1[kk]; b1.y = wp1[kk + 1];
      acc0 = __builtin_amdgcn_wmma_f32_16x16x4_f32(
          false, a, false, b0, (short)0, acc0, false, false);
      acc1 = __builtin_amdgcn_wmma_f32_16x16x4_f32(
          false, a, false, b1, (short)0, acc1, false, false);
    }
  }

  // epilogue: two 16x16 subtiles per wave
  const int tileM = blockM + wm * 16;
  const int nbase = blockN + wn * 32;
#pragma unroll
  for (int s = 0; s < 2; ++s) {
    const int n = nbase + s * 16 + idx;
    const v8f& acc = s ? acc1 : acc0;
    if (n < N) {
      const float bv = bias[n];
#pragma unroll
      for (int j = 0; j < 8; ++j) {
        const int mr = tileM + j + 8 * half;
        if (mr < M) {
          float v = acc[j] + bv;
          if (RELU) v = fmaxf(v, 0.0f);
          C[(size_t)mr * (size_t)N + n] = v;
        }
      }
    }
  }
}

// ---------------------------------------------------------------------------
// Channel attention: S=120 tokens, heads=5, hd=5.  One block per (batch, head).
// qkv layout: row (s*Bc + bb), 75 cols = [q(25) | k(25) | v(25)], head h at h*5+d.
// Online softmax (matches reference's max-subtracted softmax).
// ---------------------------------------------------------------------------
__global__ __launch_bounds__(128) void ch_attn_kernel(
    const float* __restrict__ qkv, float* __restrict__ o, int Bc) {
  const int bb = blockIdx.x / 5;
  const int h  = blockIdx.x % 5;
  __shared__ float ks[120][5];
  __shared__ float vs[120][5];
  const int t = threadIdx.x;
  if (t < 120) {
    const float* row = qkv + ((size_t)t * Bc + bb) * 75;
#pragma unroll
    for (int d = 0; d < 5; ++d) {
      ks[t][d] = row[25 + h * 5 + d];
      vs[t][d] = row[50 + h * 5 + d];
    }
  }
  __syncthreads();
  if (t >= 120) return;

  const float* qrow = qkv + ((size_t)t * Bc + bb) * 75;
  float q[5];
#pragma unroll
  for (int d = 0; d < 5; ++d) q[d] = qrow[h * 5 + d];

  const float scale = 0.44721359549995793f;  // 1/sqrt(5)
  float mx = -3.0e38f, sum = 0.0f;
  float acc[5] = {0.f, 0.f, 0.f, 0.f, 0.f};
  for (int u = 0; u < 120; ++u) {
    float s = 0.0f;
#pragma unroll
    for (int d = 0; d < 5; ++d) s += q[d] * ks[u][d];
    s *= scale;
    const float mnew = fmaxf(mx, s);
    const float corr = expf(mx - mnew);
    const float e    = expf(s - mnew);
    sum = sum * corr + e;
#pragma unroll
    for (int d = 0; d < 5; ++d) acc[d] = acc[d] * corr + e * vs[u][d];
    mx = mnew;
  }
  const float rs = 1.0f / sum;
  float* orow = o + ((size_t)t * Bc + bb) * 25;
#pragma unroll
  for (int d = 0; d < 5; ++d) orow[h * 5 + d] = acc[d] * rs;
}

// ---------------------------------------------------------------------------
// Fused residual-add + LayerNorm over last dim E (E <= 128).
// One wave32 per row; 8 rows per 256-thread block.
// ---------------------------------------------------------------------------
__global__ __launch_bounds__(256) void add_ln_kernel(
    const float* __restrict__ a, const float* __restrict__ x,
    const float* __restrict__ g, const float* __restrict__ b,
    float* __restrict__ out, int rows, int E) {
  const int wave = threadIdx.x >> 5;
  const int lane = threadIdx.x & 31;
  const int row = blockIdx.x * 8 + wave;
  if (row >= rows) return;
  const size_t base = (size_t)row * (size_t)E;
  float vals[4];
  float s = 0.0f;
#pragma unroll
  for (int j = 0; j < 4; ++j) {
    const int i = lane + 32 * j;
    float v = 0.0f;
    if (i < E) v = a[base + i] + x[base + i];
    vals[j] = v;
    s += v;
  }
#pragma unroll
  for (int off = 16; off > 0; off >>= 1) s += __shfl_xor(s, off, 32);
  const float mean = s / (float)E;
  float vsum = 0.0f;
#pragma unroll
  for (int j = 0; j < 4; ++j) {
    const int i = lane + 32 * j;
    if (i < E) {
      const float d = vals[j] - mean;
      vsum += d * d;
    }
  }
#pragma unroll
  for (int off = 16; off > 0; off >>= 1) vsum += __shfl_xor(vsum, off, 32);
  const float inv = rsqrtf(vsum / (float)E + LN_EPS);
#pragma unroll
  for (int j = 0; j < 4; ++j) {
    const int i = lane + 32 * j;
    if (i < E) out[base + i] = (vals[j] - mean) * inv * g[i] + b[i];
  }
}

// ---------------------------------------------------------------------------
// Layout helpers
// ---------------------------------------------------------------------------
__global__ void ch_gather_kernel(const float* __restrict__ x, float* __restrict__ out,
                                 int b0, int Bc) {
  const int total = 120 * Bc * 25;
  const int i = blockIdx.x * blockDim.x + threadIdx.x;
  if (i >= total) return;
  const int p = i % 25;
  const int r = i / 25;
  const int bb = r % Bc;
  const int c = r / Bc;
  out[i] = x[(((size_t)(b0 + bb)) * 120 + c) * 25 + p];
}

__global__ void ch_scatter_kernel(const float* __restrict__ cur, float* __restrict__ combined,
                                  int b0, int Bc) {
  const int total = 120 * Bc * 25;
  const int i = blockIdx.x * blockDim.x + threadIdx.x;
  if (i >= total) return;
  const int p = i % 25;
  const int r = i / 25;
  const int bb = r % Bc;
  const int c = r / Bc;
  combined[((size_t)(b0 + bb)) * 3120 + c * 25 + p] = cur[i];
}

__global__ void sp_gather_kernel(const float* __restrict__ x, float* __restrict__ out, int p) {
  const int i = blockIdx.x * blockDim.x + threadIdx.x;  // b*120 + c
  if (i >= 2048 * 120) return;
  out[i] = x[(size_t)i * 25 + p];
}

__global__ void mean_acc_kernel(const float* __restrict__ xin, float* __restrict__ acc, int n) {
  const int i = blockIdx.x * blockDim.x + threadIdx.x;
  if (i < n) acc[i] += xin[i] * 0.04f;  // 1/25
}

__global__ void sp_to_combined_kernel(const float* __restrict__ sp_mean,
                                      float* __restrict__ combined) {
  const int i = blockIdx.x * blockDim.x + threadIdx.x;
  if (i >= 2048 * 120) return;
  const int b = i / 120;
  const int c = i % 120;
  combined[(size_t)b * 3120 + 3000 + c] = sp_mean[i];
}

__global__ void zero_kernel(float* __restrict__ p, int n) {
  const int i = blockIdx.x * blockDim.x + threadIdx.x;
  if (i < n) p[i] = 0.0f;
}

// ---------------------------------------------------------------------------
// Host orchestration
// ---------------------------------------------------------------------------
struct BlockP {
  const float *in_w, *in_b, *out_w, *out_b, *g1, *b1, *g2, *b2, *w1, *bf1, *w2, *bf2;
};

static void launch_gemm(const float* A, const float* W, const float* bias, float* C,
                        int M, int N, int K, bool relu, hipStream_t s) {
  dim3 grid((unsigned)((M + 63) / 64), (unsigned)((N + 63) / 64));
  const bool vec = (K & 3) == 0;
  if (relu) {
    if (vec) wmma_gemm_kernel<1, 1><<<grid, 256, 0, s>>>(A, W, bias, C, M, N, K);
    else     wmma_gemm_kernel<1, 0><<<grid, 256, 0, s>>>(A, W, bias, C, M, N, K);
  } else {
    if (vec) wmma_gemm_kernel<0, 1><<<grid, 256, 0, s>>>(A, W, bias, C, M, N, K);
    else     wmma_gemm_kernel<0, 0><<<grid, 256, 0, s>>>(A, W, bias, C, M, N, K);
  }
}

extern "C" void kernel_launch(void* const* d_in, const int* in_sizes, int n_in,
                              void* d_out, int out_size, void* d_ws, size_t ws_size,
                              hipStream_t stream) {
  (void)n_in; (void)out_size;
  const int B = 2048, C = 120, P = 25;

  // ---- input index mapping (handles insertion-order and pytree-alphabetical) ----
  int idx_x, ch_base, sp_base, idx_fc1w, idx_fc1b, idx_fc2w, idx_fc2b;
  if (in_sizes[0] == B * C * P) {  // top-level insertion order: x first
    idx_x = 0; ch_base = 1; sp_base = 25;
    idx_fc1w = 49; idx_fc1b = 50; idx_fc2w = 51; idx_fc2b = 52;
  } else {  // alphabetical: ch_params, fc1_b, fc1_w, fc2_b, fc2_w, sp_params, x
    ch_base = 0; idx_fc1b = 24; idx_fc1w = 25; idx_fc2b = 26; idx_fc2w = 27;
    sp_base = 28; idx_x = 52;
  }
  auto F = [&](int i) -> const float* { return (const float*)d_in[i]; };
  auto getBlock = [&](int base) -> BlockP {
    BlockP Pp;
    if (in_sizes[base] > 1000) {  // insertion order inside dict (in_w first)
      Pp.in_w = F(base + 0);  Pp.in_b = F(base + 1);
      Pp.out_w = F(base + 2); Pp.out_b = F(base + 3);
      Pp.g1 = F(base + 4);    Pp.b1 = F(base + 5);
      Pp.g2 = F(base + 6);    Pp.b2 = F(base + 7);
      Pp.w1 = F(base + 8);    Pp.bf1 = F(base + 9);
      Pp.w2 = F(base + 10);   Pp.bf2 = F(base + 11);
    } else {  // alphabetical: b1,b2,bf1,bf2,g1,g2,in_b,in_w,out_b,out_w,w1,w2
      Pp.b1 = F(base + 0);    Pp.b2 = F(base + 1);
      Pp.bf1 = F(base + 2);   Pp.bf2 = F(base + 3);
      Pp.g1 = F(base + 4);    Pp.g2 = F(base + 5);
      Pp.in_b = F(base + 6);  Pp.in_w = F(base + 7);
      Pp.out_b = F(base + 8); Pp.out_w = F(base + 9);
      Pp.w1 = F(base + 10);   Pp.w2 = F(base + 11);
    }
    return Pp;
  };

  const float* x = F(idx_x);
  const float* fc1_w = F(idx_fc1w);
  const float* fc1_b = F(idx_fc1b);
  const float* fc2_w = F(idx_fc2w);
  const float* fc2_b = F(idx_fc2b);
  float* out = (float*)d_out;

  // ---- workspace layout (floats) ----
  const size_t SZ_COMB = (size_t)B * 3120;   // 6,389,760
  const size_t SZ_H1   = (size_t)B * 2048;   // 4,194,304
  const size_t SZ_SPM  = (size_t)B * C;      //   245,760
  const size_t SP_SCR  = 4 * SZ_SPM + (size_t)B * 480;  // spatial scratch: 1,966,080
  int Bc = 128;
  size_t scr = (size_t)120 * Bc * 275;       // ch scratch @ Bc=128: 4,224,000
  if (scr < SP_SCR) scr = SP_SCR;
  if (ws_size < (SZ_COMB + SZ_H1 + SZ_SPM + scr) * 4) {
    Bc = 32;
    scr = (size_t)120 * Bc * 275;
    if (scr < SP_SCR) scr = SP_SCR;
    if (ws_size < (SZ_COMB + SZ_H1 + SZ_SPM + scr) * 4) return;  // cannot run safely
  }
  float* ws = (float*)d_ws;
  float* combined = ws;
  float* h1       = combined + SZ_COMB;
  float* sp_mean  = h1 + SZ_H1;
  float* scratch  = sp_mean + SZ_SPM;

  zero_kernel<<<(unsigned)((SZ_SPM + 255) / 256), 256, 0, stream>>>(sp_mean, (int)SZ_SPM);

  // =========================== channel branch ===========================
  {
    const int chM = C * Bc;  // rows per chunk
    float* cur  = scratch;
    float* qkv  = cur + (size_t)chM * 25;
    float* o    = qkv + (size_t)chM * 75;
    float* ao   = o + (size_t)chM * 25;
    float* x1   = ao + (size_t)chM * 25;
    float* ffnh = x1 + (size_t)chM * 25;
    const int total = chM * 25;
    for (int b0 = 0; b0 < B; b0 += Bc) {
      ch_gather_kernel<<<(total + 255) / 256, 256, 0, stream>>>(x, cur, b0, Bc);
      for (int l = 0; l < 2; ++l) {
        BlockP Pp = getBlock(ch_base + l * 12);
        launch_gemm(cur, Pp.in_w, Pp.in_b, qkv, chM, 75, 25, false, stream);
        ch_attn_kernel<<<Bc * 5, 128, 0, stream>>>(qkv, o, Bc);
        launch_gemm(o, Pp.out_w, Pp.out_b, ao, chM, 25, 25, false, stream);
        add_ln_kernel<<<(chM + 7) / 8, 256, 0, stream>>>(ao, cur, Pp.g1, Pp.b1, x1, chM, 25);
        launch_gemm(x1, Pp.w1, Pp.bf1, ffnh, chM, 100, 25, true, stream);
        launch_gemm(ffnh, Pp.w2, Pp.bf2, ao, chM, 25, 100, false, stream);
        add_ln_kernel<<<(chM + 7) / 8, 256, 0, stream>>>(ao, x1, Pp.g2, Pp.b2, cur, chM, 25);
      }
      ch_scatter_kernel<<<(total + 255) / 256, 256, 0, stream>>>(cur, combined, b0, Bc);
    }
  }

  // =========================== spatial branch ===========================
  // seq_len == 1 => softmax is identity => attention output == V projection.
  {
    float* cur  = scratch;
    float* v    = cur + SZ_SPM;
    float* a    = v + SZ_SPM;
    float* x1   = a + SZ_SPM;
    float* ffnh = x1 + SZ_SPM;
    const int n = B * C;
    for (int p = 0; p < P; ++p) {
      sp_gather_kernel<<<(n + 255) / 256, 256, 0, stream>>>(x, cur, p);
      for (int l = 0; l < 2; ++l) {
        BlockP Pp = getBlock(sp_base + l * 12);
        // V projection only: rows 2E..3E-1 of in_w  (E = 120)
        launch_gemm(cur, Pp.in_w + (size_t)240 * 120, Pp.in_b + 240, v, B, 120, 120, false, stream);
        launch_gemm(v, Pp.out_w, Pp.out_b, a, B, 120, 120, false, stream);
        add_ln_kernel<<<(B + 7) / 8, 256, 0, stream>>>(a, cur, Pp.g1, Pp.b1, x1, B, 120);
        launch_gemm(x1, Pp.w1, Pp.bf1, ffnh, B, 480, 120, true, stream);
        launch_gemm(ffnh, Pp.w2, Pp.bf2, a, B, 120, 480, false, stream);
        add_ln_kernel<<<(B + 7) / 8, 256, 0, stream>>>(a, x1, Pp.g2, Pp.b2, cur, B, 120);
      }
      mean_acc_kernel<<<(n + 255) / 256, 256, 0, stream>>>(cur, sp_mean, n);
    }
  }

  // =========================== head ===========================
  sp_to_combined_kernel<<<(B * C + 255) / 256, 256, 0, stream>>>(sp_mean, combined);
  launch_gemm(combined, fc1_w, fc1_b, h1, B, 2048, 3120, true, stream);
  launch_gemm(h1, fc2_w, fc2_b, out, B, 20, 2048, false, stream);
}